// ResidualQuantizer_28003186770446
// MI455X (gfx1250) — compile-verified
//
#include <hip/hip_runtime.h>

typedef __attribute__((ext_vector_type(2))) float v2f;
typedef __attribute__((ext_vector_type(8))) float v8f;

#define Bn (1u << 20)          // batch
#define Dd 64                  // latent dim
#define Kc 256                 // codes per level
#define Lv 4                   // levels
#define CBS 68                 // padded LDS row stride (floats) -> conflict-free
#define WAVES 8
#define ROWS_PER_WAVE 16
#define ROWS_PER_BLOCK (WAVES * ROWS_PER_WAVE)

// Dynamic LDS layout (floats):
//   sCB    : Kc*CBS                  (padded codebook level, 69632 B)
//   sStage : WAVES*ROWS_PER_WAVE*CBS (per-wave z/q staging, 34816 B)
//   sNorm  : Kc                      (-0.5*|c|^2 per code, 1024 B)
//   sIdx   : WAVES*4 uint            (packed winning indices, 128 B)
#define SMEM_FLOATS (Kc * CBS + WAVES * ROWS_PER_WAVE * CBS + Kc)
#define SMEM_BYTES (SMEM_FLOATS * 4 + WAVES * 16)

__global__ __launch_bounds__(256) void rq_wmma_kernel(
    const float* __restrict__ z, const float* __restrict__ cb,
    float* __restrict__ out_qst, float* __restrict__ out_idx,
    float* __restrict__ out_q) {
  extern __shared__ float smem[];
  float* sCB = smem;                                    // codebook level
  float* sStage = smem + Kc * CBS;                      // staging
  float* sNorm = sStage + WAVES * ROWS_PER_WAVE * CBS;  // -0.5*|c|^2
  unsigned int* sIdx = (unsigned int*)(sNorm + Kc);     // packed byte indices

  const int tid = threadIdx.x;
  const int wave = tid >> 5;
  const int lane = tid & 31;
  const int row = lane & 15;         // local row (A/B fragment N or M index)
  const int off = (lane >> 4) << 1;  // K sub-offset: lanes<16 -> {0,1}, else {2,3}

  const int waveRow = blockIdx.x * ROWS_PER_BLOCK + wave * ROWS_PER_WAVE;
  float* sMyStage = sStage + wave * ROWS_PER_WAVE * CBS;

  // ---- coalesced load of this wave's 16x64 z tile into padded staging ----
  for (int it = 0; it < 8; ++it) {
    int c = it * 32 + lane;
    int r = c >> 4, c4 = (c & 15) << 2;
    float4 v = *(const float4*)(z + (size_t)(waveRow + r) * Dd + c4);
    *(float4*)(sMyStage + r * CBS + c4) = v;
  }
  __syncthreads();

  // ---- build residual A-fragments: v2f per K-chunk of 4 (split across halves) ----
  v2f res[16], quant[16];
#pragma unroll
  for (int kk = 0; kk < 16; ++kk) {
    res[kk] = *(const v2f*)(sMyStage + row * CBS + kk * 4 + off);
    quant[kk] = v2f{0.0f, 0.0f};
  }

  float idxf[Lv];

  for (int level = 0; level < Lv; ++level) {
    __syncthreads();  // protect LDS codebook + sIdx reuse across levels
    // ---- cooperative coalesced load of codebook level into padded LDS ----
    const float* cbL = cb + (size_t)level * Kc * Dd;
    for (int i = 0; i < 16; ++i) {
      int f = i * 1024 + tid * 4;  // flat float index in this level
      float4 v = *(const float4*)(cbL + f);
      *(float4*)(sCB + (f >> 6) * CBS + (f & 63)) = v;
    }
    __syncthreads();
    // ---- per-code bias: -0.5*|c|^2 ----
    {
      float s = 0.f;
#pragma unroll
      for (int i = 0; i < 16; ++i) {
        float4 v = *(const float4*)(sCB + tid * CBS + i * 4);
        s += v.x * v.x + v.y * v.y + v.z * v.z + v.w * v.w;
      }
      sNorm[tid] = -0.5f * s;
    }
    __syncthreads();

    // ---- scores = r.c - 0.5|c|^2 via chained V_WMMA_F32_16X16X4_F32 ----
    float bs[8];
    int bi[8];
#pragma unroll
    for (int j = 0; j < 8; ++j) {
      bs[j] = -3.402823466e38f;
      bi[j] = 0;
    }

#pragma unroll 2
    for (int nt = 0; nt < 16; ++nt) {
      int n = nt * 16 + row;  // this lane's N column (code index)
      float bias = sNorm[n];
      v8f acc = {bias, bias, bias, bias, bias, bias, bias, bias};
      const float* bp = sCB + n * CBS + off;
#pragma unroll
      for (int kk = 0; kk < 16; ++kk) {
        v2f b = *(const v2f*)(bp + kk * 4);
        acc = __builtin_amdgcn_wmma_f32_16x16x4_f32(
            false, res[kk], false, b, (short)0, acc, false, false);
      }
#pragma unroll
      for (int j = 0; j < 8; ++j) {
        if (acc[j] > bs[j]) {  // strict > keeps lowest index on tie (per lane)
          bs[j] = acc[j];
          bi[j] = n;
        }
      }
    }

    // ---- cross-lane argmax within each 16-lane column group (wave32) ----
#pragma unroll
    for (int j = 0; j < 8; ++j) {
#pragma unroll
      for (int m = 1; m <= 8; m <<= 1) {
        float os = __shfl_xor(bs[j], m);
        int oi = __shfl_xor(bi[j], m);
        if (os > bs[j] || (os == bs[j] && oi < bi[j])) {
          bs[j] = os;
          bi[j] = oi;
        }
      }
    }

    // ---- lanes 0 / 16 publish rows 0-7 / 8-15 as packed bytes ----
    if ((lane & 15) == 0) {
      unsigned p0 = (unsigned)bi[0] | ((unsigned)bi[1] << 8) |
                    ((unsigned)bi[2] << 16) | ((unsigned)bi[3] << 24);
      unsigned p1 = (unsigned)bi[4] | ((unsigned)bi[5] << 8) |
                    ((unsigned)bi[6] << 16) | ((unsigned)bi[7] << 24);
      sIdx[wave * 4 + (lane >> 4) * 2 + 0] = p0;
      sIdx[wave * 4 + (lane >> 4) * 2 + 1] = p1;
    }
    __syncthreads();
    int myIdx = (int)((const unsigned char*)sIdx)[wave * 16 + row];
    idxf[level] = (float)myIdx;

    // ---- gather chosen code, update residual + accumulator fragments ----
    const float* cp = sCB + myIdx * CBS + off;
#pragma unroll
    for (int kk = 0; kk < 16; ++kk) {
      v2f cv = *(const v2f*)(cp + kk * 4);
      res[kk] -= cv;
      quant[kk] += cv;
    }
  }

  __syncthreads();
  // ---- restage quantized tile, then coalesced float4 stores (2 copies) ----
#pragma unroll
  for (int kk = 0; kk < 16; ++kk) {
    *(v2f*)(sMyStage + row * CBS + kk * 4 + off) = quant[kk];
  }
  __syncthreads();

  for (int it = 0; it < 8; ++it) {
    int c = it * 32 + lane;
    int r = c >> 4, c4 = (c & 15) << 2;
    float4 v = *(const float4*)(sMyStage + r * CBS + c4);
    size_t g = (size_t)(waveRow + r) * Dd + c4;
    *(float4*)(out_qst + g) = v;
    *(float4*)(out_q + g) = v;
  }
  if (lane < 16) {
    float4 iv = make_float4(idxf[0], idxf[1], idxf[2], idxf[3]);
    *(float4*)(out_idx + (size_t)(waveRow + lane) * Lv) = iv;
  }
}

extern "C" void kernel_launch(void* const* d_in, const int* in_sizes, int n_in,
                              void* d_out, int out_size, void* d_ws,
                              size_t ws_size, hipStream_t stream) {
  (void)in_sizes; (void)n_in; (void)d_ws; (void)ws_size; (void)out_size;
  const float* z = (const float*)d_in[0];
  const float* cb = (const float*)d_in[1];
  float* out = (float*)d_out;
  float* qst = out;                         // (B, 64)
  float* idx = out + (size_t)Bn * Dd;       // (B, 4) as float
  float* q = idx + (size_t)Bn * Lv;         // (B, 64)

  // >64KB dynamic LDS: allowed on gfx1250 (320KB/WGP); request attribute.
  static_assert(SMEM_BYTES < 160 * 1024, "LDS budget");
  hipFuncSetAttribute((const void*)rq_wmma_kernel,
                      hipFuncAttributeMaxDynamicSharedMemorySize, SMEM_BYTES);

  dim3 grid(Bn / ROWS_PER_BLOCK);
  dim3 block(256);
  rq_wmma_kernel<<<grid, block, SMEM_BYTES, stream>>>(z, cb, qst, idx, q);
}